// GCN_17188459118832
// MI455X (gfx1250) — compile-verified
//
#include <hip/hip_runtime.h>

// GCN 2-layer forward for MI455X (gfx1250, wave32).
// GEMMs use native fp32 WMMA: v_wmma_f32_16x16x4_f32 (full precision).
// Scatter phases use native f32 global atomics (L2-resident working set).

typedef __attribute__((ext_vector_type(2))) float v2f;
typedef __attribute__((ext_vector_type(8))) float v8f;

#define N_NODES 100000
#define N_EDGES 3200000
#define IN_FEAT 128
#define HIDDEN  128
#define CLASSES 32

// ---------------------------------------------------------------------------
// Init: deg <- 1.0 (self loop), agg1 <- 0, out <- 0
// ---------------------------------------------------------------------------
__global__ void init_kernel(float* __restrict__ deg,
                            float* __restrict__ agg1,
                            float* __restrict__ out) {
    int i = blockIdx.x * blockDim.x + threadIdx.x;
    if (i < N_NODES) deg[i] = 1.0f;
    if (i < N_NODES * CLASSES) out[i] = 0.0f;
    if (i < N_NODES * IN_FEAT) agg1[i] = 0.0f;
}

// ---------------------------------------------------------------------------
// Degree accumulation over destination nodes
// ---------------------------------------------------------------------------
__global__ void degree_kernel(const long long* __restrict__ ei,
                              float* __restrict__ deg) {
    int e = blockIdx.x * blockDim.x + threadIdx.x;
    if (e < N_EDGES) {
        int d = (int)ei[N_EDGES + e];
        unsafeAtomicAdd(&deg[d], 1.0f);
    }
}

__global__ void rsqrt_kernel(float* __restrict__ dinv) {
    int i = blockIdx.x * blockDim.x + threadIdx.x;
    if (i < N_NODES) dinv[i] = rsqrtf(dinv[i]);   // deg >= 1 always
}

// ---------------------------------------------------------------------------
// GEMM1: h[N,128] = x[N,128] @ W1[128,128] via v_wmma_f32_16x16x4_f32
// Block: 256 threads = 8 waves; 16-node tile; each wave -> one 16x16 C tile.
// ---------------------------------------------------------------------------
__global__ __launch_bounds__(256) void gemm1_kernel(const float* __restrict__ x,
                                                    const float* __restrict__ W,
                                                    float* __restrict__ h) {
    __shared__ float xs[16 * 130];        // 16 rows, padded stride 130
    const int n0 = blockIdx.x * 16;

    // Stage 16x128 tile of x into LDS (8 floats per thread).
    {
        int t   = threadIdx.x;
        int row = t >> 4;                 // 0..15
        int c0  = (t & 15) * 8;           // 0..120
        const float* gp = x + (size_t)(n0 + row) * IN_FEAT + c0;
        float4 a = *(const float4*)(gp);
        float4 b = *(const float4*)(gp + 4);
        float* lp = xs + row * 130 + c0;
        lp[0] = a.x; lp[1] = a.y; lp[2] = a.z; lp[3] = a.w;
        lp[4] = b.x; lp[5] = b.y; lp[6] = b.z; lp[7] = b.w;
    }
    __syncthreads();

    const int wave = threadIdx.x >> 5;
    const int lane = threadIdx.x & 31;
    const int half = lane >> 4;           // lanes 16-31 hold K+2,K+3
    const int m    = lane & 15;
    const int cb   = wave * 16;           // this wave's 16 output columns

    v8f acc = {};
#pragma unroll
    for (int k0 = 0; k0 < IN_FEAT; k0 += 4) {
        const int ka = k0 + 2 * half;
        // A fragment (16x4): lane holds A[m][ka], A[m][ka+1]
        v2f a; a.x = xs[m * 130 + ka];
               a.y = xs[m * 130 + ka + 1];
        // B fragment (4x16): lane holds W[ka][cb+m], W[ka+1][cb+m]
        v2f b; b.x = W[(size_t)ka       * HIDDEN + cb + m];
               b.y = W[(size_t)(ka + 1) * HIDDEN + cb + m];
        acc = __builtin_amdgcn_wmma_f32_16x16x4_f32(
            false, a, false, b, (short)0, acc, false, false);
    }

    // C/D layout: VGPR r -> row r (lanes 0-15) or r+8 (lanes 16-31), col = m.
#pragma unroll
    for (int r = 0; r < 8; ++r) {
        int row = r + 8 * half;
        h[(size_t)(n0 + row) * HIDDEN + cb + m] = acc[r];
    }
}

// ---------------------------------------------------------------------------
// GEMM2: t2[N,32] = h2[N,128] @ W2[128,32]
// Block: 128 threads = 4 waves = 2 row-tiles x 2 col-tiles; 32-node tile.
// ---------------------------------------------------------------------------
__global__ __launch_bounds__(128) void gemm2_kernel(const float* __restrict__ h2,
                                                    const float* __restrict__ W,
                                                    float* __restrict__ t2) {
    __shared__ float hs[32 * 130];
    const int n0 = blockIdx.x * 32;

    // Stage 32x128 tile of h2 into LDS (32 floats per thread).
    {
        int t   = threadIdx.x;
        int row = t >> 2;                 // 0..31
        int c0  = (t & 3) * 32;           // 0,32,64,96
        const float* gp = h2 + (size_t)(n0 + row) * HIDDEN + c0;
        float* lp = hs + row * 130 + c0;
#pragma unroll
        for (int j = 0; j < 32; j += 4) {
            float4 a = *(const float4*)(gp + j);
            lp[j + 0] = a.x; lp[j + 1] = a.y; lp[j + 2] = a.z; lp[j + 3] = a.w;
        }
    }
    __syncthreads();

    const int wave = threadIdx.x >> 5;    // 0..3
    const int lane = threadIdx.x & 31;
    const int half = lane >> 4;
    const int m    = lane & 15;
    const int rt   = (wave >> 1) * 16;    // row tile base 0 / 16
    const int cb   = (wave & 1) * 16;     // col base 0 / 16

    v8f acc = {};
#pragma unroll
    for (int k0 = 0; k0 < HIDDEN; k0 += 4) {
        const int ka = k0 + 2 * half;
        v2f a; a.x = hs[(rt + m) * 130 + ka];
               a.y = hs[(rt + m) * 130 + ka + 1];
        v2f b; b.x = W[(size_t)ka       * CLASSES + cb + m];
               b.y = W[(size_t)(ka + 1) * CLASSES + cb + m];
        acc = __builtin_amdgcn_wmma_f32_16x16x4_f32(
            false, a, false, b, (short)0, acc, false, false);
    }

#pragma unroll
    for (int r = 0; r < 8; ++r) {
        int row = rt + r + 8 * half;
        t2[(size_t)(n0 + row) * CLASSES + cb + m] = acc[r];
    }
}

// ---------------------------------------------------------------------------
// Edge scatter, 128 features: 1 edge per wave, float4 per lane.
// ---------------------------------------------------------------------------
__global__ __launch_bounds__(256) void scatter128_kernel(const long long* __restrict__ ei,
                                                         const float* __restrict__ h,
                                                         const float* __restrict__ dinv,
                                                         float* __restrict__ agg) {
    long long gid = (long long)blockIdx.x * blockDim.x + threadIdx.x;
    long long e   = gid >> 5;
    if (e >= N_EDGES) return;
    const int lane = threadIdx.x & 31;
    long long s = ei[e];
    long long d = ei[N_EDGES + e];
    float w = dinv[s] * dinv[d];
    float4 v = *(const float4*)(h + s * IN_FEAT + lane * 4);
    float* p = agg + d * IN_FEAT + lane * 4;
    unsafeAtomicAdd(p + 0, v.x * w);
    unsafeAtomicAdd(p + 1, v.y * w);
    unsafeAtomicAdd(p + 2, v.z * w);
    unsafeAtomicAdd(p + 3, v.w * w);
}

// ---------------------------------------------------------------------------
// Edge scatter, 32 features: 1 edge per 8 lanes, float4 per lane.
// ---------------------------------------------------------------------------
__global__ __launch_bounds__(256) void scatter32_kernel(const long long* __restrict__ ei,
                                                        const float* __restrict__ t2,
                                                        const float* __restrict__ dinv,
                                                        float* __restrict__ out) {
    long long gid = (long long)blockIdx.x * blockDim.x + threadIdx.x;
    long long e   = gid >> 3;
    if (e >= N_EDGES) return;
    const int sub = threadIdx.x & 7;
    long long s = ei[e];
    long long d = ei[N_EDGES + e];
    float w = dinv[s] * dinv[d];
    float4 v = *(const float4*)(t2 + s * CLASSES + sub * 4);
    float* p = out + d * CLASSES + sub * 4;
    unsafeAtomicAdd(p + 0, v.x * w);
    unsafeAtomicAdd(p + 1, v.y * w);
    unsafeAtomicAdd(p + 2, v.z * w);
    unsafeAtomicAdd(p + 3, v.w * w);
}

// ---------------------------------------------------------------------------
// Finalize layer 1: h2 = relu(agg1 + h*dinv^2 + b1), written in place to agg1.
// ---------------------------------------------------------------------------
__global__ void finalize1_kernel(const float* __restrict__ h,
                                 const float* __restrict__ dinv,
                                 const float* __restrict__ b1,
                                 float* __restrict__ agg1) {
    int i = blockIdx.x * blockDim.x + threadIdx.x;      // over N*128/4
    if (i >= N_NODES * IN_FEAT / 4) return;
    int n  = i >> 5;                                    // /32 (=128/4)
    int f4 = (i & 31) * 4;
    float di = dinv[n];
    float w  = di * di;
    float4 a  = ((const float4*)agg1)[i];
    float4 hv = ((const float4*)h)[i];
    float4 bv = *(const float4*)(b1 + f4);
    float4 r;
    r.x = fmaxf(a.x + hv.x * w + bv.x, 0.0f);
    r.y = fmaxf(a.y + hv.y * w + bv.y, 0.0f);
    r.z = fmaxf(a.z + hv.z * w + bv.z, 0.0f);
    r.w = fmaxf(a.w + hv.w * w + bv.w, 0.0f);
    ((float4*)agg1)[i] = r;
}

// ---------------------------------------------------------------------------
// Finalize layer 2: out += t2*dinv^2 + b2   (out already holds scatter sum)
// ---------------------------------------------------------------------------
__global__ void finalize2_kernel(const float* __restrict__ t2,
                                 const float* __restrict__ dinv,
                                 const float* __restrict__ b2,
                                 float* __restrict__ out) {
    int i = blockIdx.x * blockDim.x + threadIdx.x;      // over N*32/4
    if (i >= N_NODES * CLASSES / 4) return;
    int n  = i >> 3;                                    // /8 (=32/4)
    int f4 = (i & 7) * 4;
    float di = dinv[n];
    float w  = di * di;
    float4 o  = ((const float4*)out)[i];
    float4 tv = ((const float4*)t2)[i];
    float4 bv = *(const float4*)(b2 + f4);
    o.x += tv.x * w + bv.x;
    o.y += tv.y * w + bv.y;
    o.z += tv.z * w + bv.z;
    o.w += tv.w * w + bv.w;
    ((float4*)out)[i] = o;
}

// ---------------------------------------------------------------------------
// Launcher. Inputs: x, edge_index(int64), W1, b1, W2, b2. Output: [N,32] f32.
// Workspace layout (floats): dinv[N] | h[N*128] (reused as t2) | agg1[N*128]
// total ~103 MB.
// ---------------------------------------------------------------------------
extern "C" void kernel_launch(void* const* d_in, const int* in_sizes, int n_in,
                              void* d_out, int out_size, void* d_ws, size_t ws_size,
                              hipStream_t stream) {
    const float*     x  = (const float*)d_in[0];
    const long long* ei = (const long long*)d_in[1];   // int64 per reference
    const float*     W1 = (const float*)d_in[2];
    const float*     b1 = (const float*)d_in[3];
    const float*     W2 = (const float*)d_in[4];
    const float*     b2 = (const float*)d_in[5];
    float*           out = (float*)d_out;

    float* ws   = (float*)d_ws;
    float* dinv = ws;                                  // N floats (deg, then rsqrt)
    float* h    = ws + N_NODES;                        // N*128 ; reused as t2 later
    float* agg1 = h + (size_t)N_NODES * IN_FEAT;       // N*128 ; becomes h2 in place

    // 1) init
    init_kernel<<<(N_NODES * IN_FEAT + 255) / 256, 256, 0, stream>>>(dinv, agg1, out);
    // 2) degrees + rsqrt
    degree_kernel<<<(N_EDGES + 255) / 256, 256, 0, stream>>>(ei, dinv);
    rsqrt_kernel<<<(N_NODES + 255) / 256, 256, 0, stream>>>(dinv);
    // 3) h = x @ W1   (fp32 WMMA)
    gemm1_kernel<<<N_NODES / 16, 256, 0, stream>>>(x, W1, h);
    // 4) scatter layer 1
    scatter128_kernel<<<(int)(((long long)N_EDGES * 32 + 255) / 256), 256, 0, stream>>>(ei, h, dinv, agg1);
    // 5) h2 = relu(agg1 + self + b1)  (in place)
    finalize1_kernel<<<(N_NODES * IN_FEAT / 4 + 255) / 256, 256, 0, stream>>>(h, dinv, b1, agg1);
    // 6) t2 = h2 @ W2  (fp32 WMMA) -> reuse h buffer
    gemm2_kernel<<<N_NODES / 32, 128, 0, stream>>>(agg1, W2, h);
    // 7) scatter layer 2 directly into out
    scatter32_kernel<<<(int)(((long long)N_EDGES * 8 + 255) / 256), 256, 0, stream>>>(ei, h, dinv, out);
    // 8) out += self + b2
    finalize2_kernel<<<(N_NODES * CLASSES / 4 + 255) / 256, 256, 0, stream>>>(h, dinv, b2, out);
}